// Head_65094524338989
// MI455X (gfx1250) — compile-verified
//
#include <hip/hip_runtime.h>

// ---- problem constants -------------------------------------------------
#define BATCH     32
#define NCH       255      // 3 anchors * 85
#define HW        5776     // 76*76
#define NA        3
#define NOBJ      85
#define OUTC      84       // 4 box + 80 cls (obj folded into cls)
#define TILE_P    38       // pixels per block (5776 = 152*38)
#define TILES     152
#define LDS_S     39       // LDS row stride (odd -> gcd(39,64)=1)
#define NTHREADS  256
#define P1_ACTIVE (6 * TILE_P)   // 228 threads active in phase 1
#define NQUADS    (TILE_P * NA * OUTC / 4)   // 2394 float4 quads per tile

#define LOG2E_F   1.44269504088896340736f
#define INV_W     (1.0f / 76.0f)
// anchors / stride / W  ==  anchor_px / IMG_SIZE(608)
#define AW0 (116.0f / 608.0f)
#define AH0 ( 90.0f / 608.0f)
#define AW1 (156.0f / 608.0f)
#define AH1 (198.0f / 608.0f)
#define AW2 (373.0f / 608.0f)
#define AH2 (326.0f / 608.0f)

// ---- fast-math primitives (hardware trans ops, guarded) ---------------
#if __has_builtin(__builtin_amdgcn_exp2f)
#define FEXP2(x) __builtin_amdgcn_exp2f(x)
#else
#define FEXP2(x) exp2f(x)
#endif
#if __has_builtin(__builtin_amdgcn_rcpf)
#define FRCP(x) __builtin_amdgcn_rcpf(x)
#else
#define FRCP(x) (1.0f / (x))
#endif

__device__ __forceinline__ float fsigmoid(float v) {
    return FRCP(1.0f + FEXP2(-v * LOG2E_F));
}

// ---- CDNA5 async global->LDS path (guarded with scalar fallback) ------
#define AS1 __attribute__((address_space(1)))
#define AS3 __attribute__((address_space(3)))

#if __has_builtin(__builtin_amdgcn_global_load_async_to_lds_b32)
#define HAVE_ASYNC_LDS 1
#endif

__global__ __launch_bounds__(NTHREADS) void
yolo_head_kernel(const float* __restrict__ x, float* __restrict__ out) {
    // 255 channel rows of 38 pixels, stride 39 floats; + sigmoid(obj) cache
    __shared__ float lds[NCH * LDS_S + NA * TILE_P];
    float* sobj = &lds[NCH * LDS_S];

    const int tid  = threadIdx.x;
    const int bidx = blockIdx.x;
    const int b    = bidx / TILES;
    const int tile = bidx - b * TILES;

    const float* __restrict__ src = x + (size_t)b * NCH * HW + tile * TILE_P;

    // ---- Phase 1: stage 255 x 38 tile into LDS --------------------------
    // 228 threads cover (c0 in 0..5, p in 0..37); each walks c += 6.
    // Global reads: 32 consecutive lanes mostly span consecutive pixels of
    // one channel row -> coalesced. Async global->LDS: no VGPR round-trip.
    if (tid < P1_ACTIVE) {
        int c = tid / TILE_P;               // 0..5
        const int p = tid - c * TILE_P;     // 0..37
        const float* g = src + (size_t)c * HW + p;
        int ldsIdx = c * LDS_S + p;
        for (; c < NCH; c += 6) {
#if defined(HAVE_ASYNC_LDS)
            __builtin_amdgcn_global_load_async_to_lds_b32(
                (AS1 int*)const_cast<float*>(g),
                (AS3 int*)&lds[ldsIdx],
                /*offset=*/0, /*cpol=*/0);
#else
            lds[ldsIdx] = *g;
#endif
            g += (size_t)6 * HW;
            ldsIdx += 6 * LDS_S;
        }
    }
#if defined(HAVE_ASYNC_LDS)
#if __has_builtin(__builtin_amdgcn_s_wait_asynccnt)
    __builtin_amdgcn_s_wait_asynccnt(0);
#else
    asm volatile("s_wait_asynccnt 0" ::: "memory");
#endif
#endif
    __syncthreads();

    // ---- Phase 1.5: sigmoid(objectness) once per (anchor, pixel) --------
    if (tid < NA * TILE_P) {
        const int a = tid / TILE_P;
        const int p = tid - a * TILE_P;
        sobj[a * TILE_P + p] = fsigmoid(lds[(a * NOBJ + 4) * LDS_S + p]);
    }
    __syncthreads();

    // ---- Phase 2: one float4 quad per thread-iteration ------------------
    // Output span for this tile is one contiguous block of 38*252 floats.
    // 252 = 63 quads per pixel = 3 anchors * 21 quads; quad 0 of each
    // anchor row is exactly [x, y, w, h], quads 1..20 are pure classes.
    const size_t outBase =
        (size_t)b * HW * (NA * OUTC) + (size_t)tile * TILE_P * (NA * OUTC);
    const int hrow  = tile >> 1;                 // pix / 76
    const int wbase = (tile & 1) * TILE_P;       // pix % 76 base
    const float hf  = (float)hrow;

    for (int q = tid; q < NQUADS; q += NTHREADS) {
        const int p  = q / 63;           // pixel within tile
        const int rq = q - p * 63;       // quad within pixel: 0..62
        const int a  = rq / 21;          // anchor
        const int j4 = rq - a * 21;      // quad within anchor row: 0..20
        const int cb = a * NOBJ;         // first channel of this anchor

        float4 val;
        if (j4 == 0) {
            // [x, y, w, h]
            const int i0 = cb * LDS_S + p;
            const float v0 = lds[i0];
            const float v1 = lds[i0 + LDS_S];
            const float v2 = lds[i0 + 2 * LDS_S];
            const float v3 = lds[i0 + 3 * LDS_S];
            const float aw = (a == 0) ? AW0 : (a == 1) ? AW1 : AW2;
            const float ah = (a == 0) ? AH0 : (a == 1) ? AH1 : AH2;
            val.x = (fsigmoid(v0) + (float)(wbase + p)) * INV_W;
            val.y = (fsigmoid(v1) + hf) * INV_W;
            val.z = FEXP2(v2 * LOG2E_F) * aw;
            val.w = FEXP2(v3 * LOG2E_F) * ah;
        } else {
            // 4 class channels: j = 4*j4 .. 4*j4+3, input channel j+1
            const float so = sobj[a * TILE_P + p];
            const int i0 = (cb + 4 * j4 + 1) * LDS_S + p;
            val.x = fsigmoid(lds[i0]) * so;
            val.y = fsigmoid(lds[i0 + LDS_S]) * so;
            val.z = fsigmoid(lds[i0 + 2 * LDS_S]) * so;
            val.w = fsigmoid(lds[i0 + 3 * LDS_S]) * so;
        }
        // 16B-aligned contiguous store: outBase % 4 == 0, q*4 % 4 == 0
        *(float4*)(out + outBase + (size_t)q * 4) = val;
    }
}

extern "C" void kernel_launch(void* const* d_in, const int* in_sizes, int n_in,
                              void* d_out, int out_size, void* d_ws, size_t ws_size,
                              hipStream_t stream) {
    (void)in_sizes; (void)n_in; (void)d_ws; (void)ws_size; (void)out_size;
    const float* x = (const float*)d_in[0];
    float* out = (float*)d_out;
    dim3 grid(BATCH * TILES);   // 4864 blocks
    dim3 block(NTHREADS);
    yolo_head_kernel<<<grid, block, 0, stream>>>(x, out);
}